// GCNLayer_31928786879187
// MI455X (gfx1250) — compile-verified
//
#include <hip/hip_runtime.h>
#include <math.h>

typedef __attribute__((ext_vector_type(2))) float v2f;
typedef __attribute__((ext_vector_type(8))) float v8f;

#define NODE_F 128
#define HID 64

// Hardware float atomic add, non-returning (STOREcnt), device scope.
// Guarantees global_atomic_add_f32 instead of a possible CAS-loop expansion.
__device__ __forceinline__ void atomAddF(float* p, float v) {
  asm volatile("global_atomic_add_f32 %0, %1, off scope:SCOPE_DEV"
               :: "v"(p), "v"(v) : "memory");
}

// ---------------- degree / normalization ----------------

__global__ void k_deg_init(float* __restrict__ deg, int N) {
  int i = blockIdx.x * blockDim.x + threadIdx.x;
  if (i < N) deg[i] = 1.0f;  // self-loop contributes 1 to every node's degree
}

__global__ void k_deg_count(const long long* __restrict__ dst, float* __restrict__ deg, int E) {
  int i = blockIdx.x * blockDim.x + threadIdx.x;
  if (i < E) atomAddF(&deg[(int)dst[i]], 1.0f);
}

__global__ void k_dinv(const float* __restrict__ deg, float* __restrict__ dinv, int N) {
  int i = blockIdx.x * blockDim.x + threadIdx.x;
  if (i < N) {
    float d = deg[i];
    dinv[i] = (d > 0.0f) ? rsqrtf(d) : 0.0f;
  }
}

// ---------------- dense transform via f32 WMMA ----------------
// Block = 128 threads (4 waves). Each block computes a 16-row x 64-col tile of C = A @ W.
// A: [M, K] row-major, W: [K, 64] row-major, C: [M, 64]. M must be a multiple of 16.
template <int K>
__global__ __launch_bounds__(128) void k_gemm_wmma(const float* __restrict__ A,
                                                   const float* __restrict__ W,
                                                   float* __restrict__ C, int M) {
  __shared__ float sA[16 * K];    // 16-row A tile (contiguous in memory since lda == K)
  __shared__ float sW[K * HID];   // full weight matrix

  const int tid  = threadIdx.x;
  const int row0 = blockIdx.x * 16;

  {
    const float4* gA = (const float4*)(A + (size_t)row0 * K);
    float4* lA = (float4*)sA;
    #pragma unroll
    for (int i = tid; i < 16 * K / 4; i += 128) lA[i] = gA[i];
    const float4* gW = (const float4*)W;
    float4* lW = (float4*)sW;
    for (int i = tid; i < K * HID / 4; i += 128) lW[i] = gW[i];
  }
  __syncthreads();

  const int lane = tid & 31;
  const int wave = tid >> 5;   // 0..3 -> 16-col slice of the 64 output cols
  const int n0   = wave * 16;
  const int l15  = lane & 15;
  const int half = lane >> 4;  // lane-half selects K offset per ISA 16x4 A layout

  v8f acc = {};
  #pragma unroll
  for (int k = 0; k < K; k += 4) {
    // A 16x4: lanes 0-15 rows, VGPR0=K+2*half, VGPR1=K+2*half+1
    v2f a, b;
    a.x = sA[l15 * K + k + 2 * half];
    a.y = sA[l15 * K + k + 2 * half + 1];
    // B 4x16: row K striped across lanes; VGPR index = K%2, lane-half = K/2
    b.x = sW[(k + 2 * half + 0) * HID + n0 + l15];
    b.y = sW[(k + 2 * half + 1) * HID + n0 + l15];
    acc = __builtin_amdgcn_wmma_f32_16x16x4_f32(false, a, false, b, (short)0, acc, false, false);
  }

  // C/D 16x16 f32 layout: lane = N (0-15), VGPR r -> M = r + 8*(lane>=16)
  float* out = C + (size_t)row0 * HID + n0;
  #pragma unroll
  for (int r = 0; r < 8; ++r) {
    out[(size_t)(r + 8 * half) * HID + l15] = acc[r];
  }
}

// ---------------- aggregation ----------------

// out[i][c] = h[i][c]*dinv[i]^2 (+ bias[c] if given) — self-loop term, also inits accumulator
__global__ void k_self_init(const float* __restrict__ h, const float* __restrict__ dinv,
                            const float* __restrict__ bias, float* __restrict__ out, int N) {
  int idx = blockIdx.x * blockDim.x + threadIdx.x;
  if (idx >= N * HID) return;
  int i = idx >> 6;
  int c = idx & (HID - 1);
  float di = dinv[i];
  float v = h[idx] * di * di;
  if (bias) v += bias[c];
  out[idx] = v;
}

// One wave per edge: 32 lanes x 2 columns. Edge index is wave-uniform, so we
// readfirstlane it -> src/dst/dinv loads become scalar (SMEM) loads; only the
// h-row gather (one coalesced b64 per lane) and the atomics use vector memory.
__global__ void k_edge_agg(const float* __restrict__ h, const float* __restrict__ dinv,
                           const long long* __restrict__ src, const long long* __restrict__ dst,
                           float* __restrict__ agg, int E) {
  long long idx = (long long)blockIdx.x * blockDim.x + threadIdx.x;
  if (idx >= (long long)E * 32) return;              // whole waves pass/fail together
  int e = __builtin_amdgcn_readfirstlane((int)(idx >> 5));
  int c = ((int)idx & 31) * 2;
  int s = __builtin_amdgcn_readfirstlane((int)src[e]);
  int d = __builtin_amdgcn_readfirstlane((int)dst[e]);
  float nrm = dinv[s] * dinv[d];                      // uniform -> scalar loads
  float2 hv = *(const float2*)(h + (size_t)s * HID + c);
  atomAddF(&agg[(size_t)d * HID + c + 0], hv.x * nrm);
  atomAddF(&agg[(size_t)d * HID + c + 1], hv.y * nrm);
}

// exact-erf GELU on (agg + bias)
__global__ void k_bias_gelu(const float* __restrict__ agg, const float* __restrict__ bias,
                            float* __restrict__ out, int N) {
  int idx = blockIdx.x * blockDim.x + threadIdx.x;
  if (idx >= N * HID) return;
  float v = agg[idx] + bias[idx & (HID - 1)];
  out[idx] = 0.5f * v * (1.0f + erff(v * 0.70710678118654752440f));
}

// ---------------- launch ----------------

extern "C" void kernel_launch(void* const* d_in, const int* in_sizes, int n_in,
                              void* d_out, int out_size, void* d_ws, size_t ws_size,
                              hipStream_t stream) {
  const float*      x  = (const float*)d_in[0];
  const long long*  ei = (const long long*)d_in[1];
  const float*      W1 = (const float*)d_in[2];
  const float*      b1 = (const float*)d_in[3];
  const float*      W2 = (const float*)d_in[4];
  const float*      b2 = (const float*)d_in[5];

  const int N = in_sizes[0] / NODE_F;  // 100000 (multiple of 16)
  const int E = in_sizes[1] / 2;       // 1.6M
  const long long* srcI = ei;          // edge_index[0]
  const long long* dstI = ei + E;      // edge_index[1]

  // workspace carve-up (floats): deg[N] | dinv[N] | bufA[N*64] | bufB[N*64]  (~52 MB)
  float* deg  = (float*)d_ws;
  float* dinv = deg + N;
  float* bufA = dinv + N;
  float* bufB = bufA + (size_t)N * HID;
  float* out  = (float*)d_out;

  const int TB = 256;
  const unsigned nodeBlk  = (unsigned)((N + TB - 1) / TB);
  const unsigned edgeBlk  = (unsigned)((E + TB - 1) / TB);
  const unsigned featBlk  = (unsigned)(((size_t)N * HID + TB - 1) / TB);
  const unsigned eaggBlk  = (unsigned)(((size_t)E * 32 + TB - 1) / TB);
  const unsigned gemmBlk  = (unsigned)(N / 16);

  // normalization
  k_deg_init<<<nodeBlk, TB, 0, stream>>>(deg, N);
  k_deg_count<<<edgeBlk, TB, 0, stream>>>(dstI, deg, E);
  k_dinv<<<nodeBlk, TB, 0, stream>>>(deg, dinv, N);

  // layer 1
  k_gemm_wmma<NODE_F><<<gemmBlk, 128, 0, stream>>>(x, W1, bufA, N);
  k_self_init<<<featBlk, TB, 0, stream>>>(bufA, dinv, nullptr, bufB, N);
  k_edge_agg<<<eaggBlk, TB, 0, stream>>>(bufA, dinv, srcI, dstI, bufB, E);
  k_bias_gelu<<<featBlk, TB, 0, stream>>>(bufB, b1, bufA, N);

  // layer 2
  k_gemm_wmma<HID><<<gemmBlk, 128, 0, stream>>>(bufA, W2, bufB, N);
  k_self_init<<<featBlk, TB, 0, stream>>>(bufB, dinv, b2, out, N);
  k_edge_agg<<<eaggBlk, TB, 0, stream>>>(bufB, dinv, srcI, dstI, out, E);
}